// DiffeomorphicTransform_unit_77360950936237
// MI455X (gfx1250) — compile-verified
//
#include <hip/hip_runtime.h>

// ---------------------------------------------------------------------------
// Diffeomorphic transform (scaling & squaring), MI455X / gfx1250.
// Gather-bound: ~2.8 GB traffic over 7 steps -> ~120us floor @ 23.3 TB/s.
// No matrix ops exist in this computation (WMMA inapplicable). This revision
// makes all load/store bases block-uniform (scalar) so gathers lower to
// saddr-form (SGPR64 base + 32-bit VGPR offset) instead of per-lane 64-bit
// address pairs: fewer VALU ops, fewer VGPRs, same vmem count.
// ---------------------------------------------------------------------------

namespace {
constexpr int B = 2, C = 3, D = 160, H = 160, W = 160;
constexpr int HW      = H * W;            // 25,600
constexpr int SPATIAL = D * HW;           // 4,096,000
constexpr int NVOX    = B * SPATIAL;      // 8,192,000
constexpr int CH      = SPATIAL;          // channel stride (floats)
constexpr int BF      = C * SPATIAL;      // batch stride, flow  [B,C,D,H,W]
constexpr long BG     = (long)SPATIAL*3;  // batch stride, grid  [B,D,H,W,3]
constexpr int BLKS_PER_BATCH = SPATIAL / 256;  // 16,000 exactly
}

__global__ __launch_bounds__(256) void warp_step(
    const float* __restrict__ src,   // flow (or velocity on step 1)
    const float* __restrict__ grid,  // identity sample grid [B,D,H,W,3] (x,y,z)
    float*       __restrict__ dst,   // flow out
    float scale)                     // 1/2^7 on step 1, else 1.0
{
  // Batch index is uniform per block (scalar compare, no division): blocks
  // [0,16000) -> b=0, [16000,32000) -> b=1.
  const int blk = (int)blockIdx.x;
  const int b   = (blk >= BLKS_PER_BATCH) ? 1 : 0;
  const int sp  = (blk - b * BLKS_PER_BATCH) * 256 + (int)threadIdx.x;

  // Uniform (SGPR-resident) base pointers; per-lane part is a 32-bit offset.
  const float* __restrict__ s0 = src + (size_t)b * BF;       // channel 0
  const float* __restrict__ s1 = s0 + CH;                    // channel 1
  const float* __restrict__ s2 = s1 + CH;                    // channel 2
  const float* __restrict__ gb = grid + (size_t)b * BG;
  float* __restrict__ d0 = dst + (size_t)b * BF;
  float* __restrict__ d1 = d0 + CH;
  float* __restrict__ d2 = d1 + CH;

  // ---- streamed reads: flow at this voxel (3 ch) + identity grid (x,y,z) ---
  const float f0 = s0[sp] * scale;
  const float f1 = s1[sp] * scale;
  const float f2 = s2[sp] * scale;
  const int gidx = sp * 3;
  const float gx = gb[gidx + 0] + f0;
  const float gy = gb[gidx + 1] + f1;
  const float gz = gb[gidx + 2] + f2;

  // gfx1250 L2 prefetch one block ahead on the streaming data (speculative;
  // dropped silently on translation failure per ISA Sec. 10.5).
  __builtin_prefetch(s2 + sp + 256,   0, 1);
  __builtin_prefetch(gb + gidx + 768, 0, 1);

  // ---- normalized [-1,1] -> voxel coords, border clamp (align_corners) -----
  float ix = (gx + 1.0f) * 0.5f * (float)(W - 1);
  float iy = (gy + 1.0f) * 0.5f * (float)(H - 1);
  float iz = (gz + 1.0f) * 0.5f * (float)(D - 1);
  ix = fminf(fmaxf(ix, 0.0f), (float)(W - 1));
  iy = fminf(fmaxf(iy, 0.0f), (float)(H - 1));
  iz = fminf(fmaxf(iz, 0.0f), (float)(D - 1));

  const float x0f = floorf(ix), y0f = floorf(iy), z0f = floorf(iz);
  const float fx = ix - x0f, fy = iy - y0f, fz = iz - z0f;
  const int x0 = (int)x0f, y0 = (int)y0f, z0 = (int)z0f;
  const int x1 = min(x0 + 1, W - 1);
  const int y1 = min(y0 + 1, H - 1);
  const int z1 = min(z0 + 1, D - 1);

  const int zo0 = z0 * HW, zo1 = z1 * HW;
  const int yo0 = y0 * W,  yo1 = y1 * W;
  // 8 per-lane spatial offsets, shared across all 3 channel base pointers.
  const int i000 = zo0 + yo0 + x0, i001 = zo0 + yo0 + x1;
  const int i010 = zo0 + yo1 + x0, i011 = zo0 + yo1 + x1;
  const int i100 = zo1 + yo0 + x0, i101 = zo1 + yo0 + x1;
  const int i110 = zo1 + yo1 + x0, i111 = zo1 + yo1 + x1;

  // ---- issue all 24 gathers before any lerp math (max MLP) -----------------
  float a0 = s0[i000], a1 = s0[i001], a2 = s0[i010], a3 = s0[i011];
  float a4 = s0[i100], a5 = s0[i101], a6 = s0[i110], a7 = s0[i111];
  float b0 = s1[i000], b1 = s1[i001], b2 = s1[i010], b3 = s1[i011];
  float b4 = s1[i100], b5 = s1[i101], b6 = s1[i110], b7 = s1[i111];
  float c0_ = s2[i000], c1_ = s2[i001], c2_ = s2[i010], c3_ = s2[i011];
  float c4_ = s2[i100], c5_ = s2[i101], c6_ = s2[i110], c7_ = s2[i111];

  const float wx0 = 1.0f - fx, wy0 = 1.0f - fy, wz0 = 1.0f - fz;

  // channel 0
  {
    const float c00 = a0 * wx0 + a1 * fx;
    const float c01 = a2 * wx0 + a3 * fx;
    const float c10 = a4 * wx0 + a5 * fx;
    const float c11 = a6 * wx0 + a7 * fx;
    const float e0  = c00 * wy0 + c01 * fy;
    const float e1  = c10 * wy0 + c11 * fy;
    d0[sp] = f0 + (e0 * wz0 + e1 * fz) * scale;  // sample(s*f) == s*sample(f)
  }
  // channel 1
  {
    const float c00 = b0 * wx0 + b1 * fx;
    const float c01 = b2 * wx0 + b3 * fx;
    const float c10 = b4 * wx0 + b5 * fx;
    const float c11 = b6 * wx0 + b7 * fx;
    const float e0  = c00 * wy0 + c01 * fy;
    const float e1  = c10 * wy0 + c11 * fy;
    d1[sp] = f1 + (e0 * wz0 + e1 * fz) * scale;
  }
  // channel 2
  {
    const float c00 = c0_ * wx0 + c1_ * fx;
    const float c01 = c2_ * wx0 + c3_ * fx;
    const float c10 = c4_ * wx0 + c5_ * fx;
    const float c11 = c6_ * wx0 + c7_ * fx;
    const float e0  = c00 * wy0 + c01 * fy;
    const float e1  = c10 * wy0 + c11 * fy;
    d2[sp] = f2 + (e0 * wz0 + e1 * fz) * scale;
  }
}

extern "C" void kernel_launch(void* const* d_in, const int* in_sizes, int n_in,
                              void* d_out, int out_size, void* d_ws, size_t ws_size,
                              hipStream_t stream) {
  (void)in_sizes; (void)n_in; (void)out_size; (void)ws_size;

  const float* velocity = (const float*)d_in[0];   // [B,C,D,H,W] f32
  const float* grid     = (const float*)d_in[1];   // [B,D,H,W,3] f32
  float* out = (float*)d_out;                      // [B,C,D,H,W] f32
  float* ws  = (float*)d_ws;                       // one flow buffer: 98.3 MB

  const dim3 grd(NVOX / 256), blk(256);
  const float s0 = 1.0f / 128.0f;                  // 1 / 2^TIME_STEP

  // 7 squaring steps; odd steps write d_out so step 7 lands in d_out.
  warp_step<<<grd, blk, 0, stream>>>(velocity, grid, out, s0);   // X1 -> out
  warp_step<<<grd, blk, 0, stream>>>(out,      grid, ws,  1.f);  // X2 -> ws
  warp_step<<<grd, blk, 0, stream>>>(ws,       grid, out, 1.f);  // X3 -> out
  warp_step<<<grd, blk, 0, stream>>>(out,      grid, ws,  1.f);  // X4 -> ws
  warp_step<<<grd, blk, 0, stream>>>(ws,       grid, out, 1.f);  // X5 -> out
  warp_step<<<grd, blk, 0, stream>>>(out,      grid, ws,  1.f);  // X6 -> ws
  warp_step<<<grd, blk, 0, stream>>>(ws,       grid, out, 1.f);  // X7 -> out
}